// HOGLayerC_32143535243483
// MI455X (gfx1250) — compile-verified
//
#include <hip/hip_runtime.h>
#include <stdint.h>

// HOG fused kernel for MI455X (gfx1250, wave32).
// Grid: (14 block-rows, 64 batches), 256 threads (8 waves) per workgroup.
// Each workgroup: stage 3x18x226 reflect-padded tile -> LDS (async-to-LDS),
// sobel+atan2+ds_add_f32 pooling -> cell histograms, WMMA f32 ones-trick for
// the 9-bin sum-of-squares, normalize + 2x2-cell block packing -> 14x108 out.
// Roofline: 38.5MB read + 5.4MB write @ 23.3TB/s ~= 1.9us floor; atan2/sqrt
// VALU work is comparable, so a fully fused single pass is the right shape.

#define NBINS   9
#define HGT     224
#define WID     224
#define HC      28            // cells per row
#define HB      14            // blocks per row
#define STRIDE  232           // floats per staged LDS row (16B-aligned b128 dests)
#define NROWS   18            // 16 pixel rows + 2 halo rows
#define NGROUPS (3 * 2 * HC)  // 168 live (ch, cell-row-in-wg, cell-col) groups
#define GPAD    176           // padded group count (11 WMMA chunks x 16)
#define BPAD    12            // padded bins per group (3 K=4 WMMA chunks)

typedef __attribute__((ext_vector_type(2))) float v2f;
typedef __attribute__((ext_vector_type(8))) float v8f;
typedef int v4i_gcc __attribute__((vector_size(16)));   // matches builtin param type

#define AS1 __attribute__((address_space(1)))
#define AS3 __attribute__((address_space(3)))

#if __has_builtin(__builtin_amdgcn_global_load_async_to_lds_b128) && \
    __has_builtin(__builtin_amdgcn_global_load_async_to_lds_b32)
#define HAVE_ASYNC_LDS 1
#endif

__device__ static inline void wait_async_zero() {
#ifdef HAVE_ASYNC_LDS
#if __has_builtin(__builtin_amdgcn_s_wait_asynccnt)
  __builtin_amdgcn_s_wait_asynccnt(0);
#else
  asm volatile("s_wait_asynccnt 0" ::: "memory");
#endif
#endif
}

__global__ __launch_bounds__(256) void hog_fused_kernel(
    const float* __restrict__ x, float* __restrict__ out, int batch) {
  __shared__ float smIn[3 * NROWS * STRIDE];   // 50112 B staged input tile
  __shared__ float smHist[GPAD * BPAD];        // 8448 B padded cell histograms
  __shared__ float smRDen[GPAD];               // 704 B reciprocal L2 denoms

  const int tid = threadIdx.x;
  const int b   = blockIdx.y;
  const int hb  = blockIdx.x;          // block row: pixel rows [hb*16, hb*16+16)
  const int y0  = hb * 16;

  // ---------- Stage input tile into LDS (reflect padding resolved here) ----
  // Interior: 4-float (b128) chunks, 3*18*56 = 3024 chunks.
  for (int t = tid; t < 3 * NROWS * 56; t += 256) {
    int ch  = t / (NROWS * 56);
    int rem = t - ch * (NROWS * 56);
    int r   = rem / 56;
    int j   = rem - r * 56;
    int yy  = y0 - 1 + r;                       // global row with halo
    yy = (yy < 0) ? -yy : yy;                   // reflect: xp[-1] = x[1]
    yy = (yy > 223) ? (446 - yy) : yy;          // reflect: xp[224] = x[222]
    const float* g = x + (((size_t)(b * 3 + ch) * HGT + yy) * WID + 4 * j);
    float* l = &smIn[(ch * NROWS + r) * STRIDE + 4 + 4 * j];
#ifdef HAVE_ASYNC_LDS
    __builtin_amdgcn_global_load_async_to_lds_b128(
        (AS1 v4i_gcc*)g, (AS3 v4i_gcc*)l, 0, 0);
#else
    l[0] = g[0]; l[1] = g[1]; l[2] = g[2]; l[3] = g[3];
#endif
  }
  // Halo columns: left (x=-1 -> 1) at LDS col 3, right (x=224 -> 222) at col 228.
  for (int t = tid; t < 2 * 3 * NROWS; t += 256) {
    int side = t / (3 * NROWS);
    int rem  = t - side * (3 * NROWS);
    int ch   = rem / NROWS;
    int r    = rem - ch * NROWS;
    int yy   = y0 - 1 + r;
    yy = (yy < 0) ? -yy : yy;
    yy = (yy > 223) ? (446 - yy) : yy;
    const float* g = x + (((size_t)(b * 3 + ch) * HGT + yy) * WID + (side ? 222 : 1));
    float* l = &smIn[(ch * NROWS + r) * STRIDE + (side ? 228 : 3)];
#ifdef HAVE_ASYNC_LDS
    __builtin_amdgcn_global_load_async_to_lds_b32((AS1 int*)g, (AS3 int*)l, 0, 0);
#else
    l[0] = g[0];
#endif
  }
  // Zero padded histograms while async copies fly (disjoint LDS region).
  // Padding (bins 9..11, groups 168..175) stays zero -> exact f32 sums.
  for (int t = tid; t < GPAD * BPAD; t += 256) smHist[t] = 0.0f;
  wait_async_zero();
  __syncthreads();

  // ---------- Sobel + magnitude/bin + 8x8 pooling via LDS float atomics ----
  const float kBinScale = 2.8647889756541161f;  // NBINS / pi
  for (int t = tid; t < 3 * 16 * WID; t += 256) {
    int ch  = t / (16 * WID);
    int rem = t - ch * (16 * WID);
    int ry  = rem / WID;                 // 0..15 within block row
    int xx  = rem - ry * WID;            // 0..223
    const float* p = &smIn[(ch * NROWS + ry + 1) * STRIDE + 4 + xx];
    float tl = p[-STRIDE - 1], tc = p[-STRIDE], tr = p[-STRIDE + 1];
    float ml = p[-1],                                mr = p[1];
    float bl = p[STRIDE - 1],  bc = p[STRIDE],  br = p[STRIDE + 1];
    // cross-correlation with wx=[[1,0,-1],[2,0,-2],[1,0,-1]], wy = wx^T
    float gxv = (tl - tr) + 2.0f * (ml - mr) + (bl - br);
    float gyv = (tl - bl) + 2.0f * (tc - bc) + (tr - br);
    float mag = sqrtf(gxv * gxv + gyv * gyv);
    float ph  = atan2f(gxv, gyv) * kBinScale;    // (-9, 9]
    int k = (int)floorf(ph);
    k %= NBINS; if (k < 0) k += NBINS;           // python-mod into [0,9)
    int grp = (ch * 2 + (ry >> 3)) * HC + (xx >> 3);
    atomicAdd(&smHist[grp * BPAD + k], mag);     // ds_add_f32
  }
  __syncthreads();

  // ---------- L2 denominators over 9 bins via V_WMMA_F32_16X16X4_F32 -------
  // D = A x ones(4x16) + C gives per-row (per-group) sums; A holds squared
  // bin values. 3 chained WMMAs cover K=0..11 (padding is zero -> exact).
  // A layout (ISA 7.12.2): lanes 0-15 hold M=lane with VGPR0=K0,VGPR1=K1;
  // lanes 16-31 hold M=lane-16 with VGPR0=K2,VGPR1=K3. B=ones is layout-free.
  // Loads are unconditional aligned float2 (ds_load_b64): no EXEC churn.
  {
    const int wave = tid >> 5;
    const int lane = tid & 31;
    const bool hi  = lane >= 16;
    for (int g = wave; g < GPAD / 16; g += 8) {   // 11 chunks over 8 waves
      int m = g * 16 + (lane & 15);               // m < 176 always
      v8f acc = {0.f, 0.f, 0.f, 0.f, 0.f, 0.f, 0.f, 0.f};
      v2f ones; ones.x = 1.0f; ones.y = 1.0f;
#pragma unroll
      for (int kb = 0; kb < BPAD; kb += 4) {
        int k0 = kb + (hi ? 2 : 0);               // even -> 8B-aligned pair
        v2f h = *(const v2f*)&smHist[m * BPAD + k0];
        v2f A = h * h;
        acc = __builtin_amdgcn_wmma_f32_16x16x4_f32(
            false, A, false, ones, (short)0, acc, false, false);
      }
      // D[m][n] identical across n; lane 0 holds M=0..7 in acc[0..7],
      // lane 16 holds M=8..15 in acc[0..7] (C/D layout, ISA 7.12.2).
      if ((lane & 15) == 0) {
        int mbase = g * 16 + (hi ? 8 : 0);
#pragma unroll
        for (int i2 = 0; i2 < 8; ++i2) {
          float den = sqrtf(acc[i2]);
          den = (den > 1e-12f) ? den : 1e-12f;
          smRDen[mbase + i2] = 1.0f / den;        // padded groups harmless
        }
      }
    }
  }
  __syncthreads();

  // ---------- Normalize + 2x2-cell block packing + global store ------------
  // hog[b, hb*14+wb, ((c*9+k)*2+i)*2+j] = hist[c, i, wb*2+j, k] * rden
  for (int t = tid; t < HB * 108; t += 256) {
    int blk = t / 108;                 // wb
    int r   = t - blk * 108;
    int j   = r & 1;                   // cell col within block
    int i2  = (r >> 1) & 1;            // cell row within block
    int cb  = r >> 2;                  // 0..26 = c*9+k
    int ch  = cb / NBINS;
    int k   = cb - ch * NBINS;
    int xc  = blk * 2 + j;
    int grp = (ch * 2 + i2) * HC + xc;
    float v = smHist[grp * BPAD + k] * smRDen[grp];
    out[((size_t)b * (HB * HB) + (size_t)(hb * HB + blk)) * 108 + r] = v;
  }
}

extern "C" void kernel_launch(void* const* d_in, const int* in_sizes, int n_in,
                              void* d_out, int out_size, void* d_ws, size_t ws_size,
                              hipStream_t stream) {
  const float* x = (const float*)d_in[0];
  float* out = (float*)d_out;
  int batch = in_sizes[0] / (3 * HGT * WID);   // 64
  dim3 grid(HB, batch);
  hog_fused_kernel<<<grid, 256, 0, stream>>>(x, out, batch);
}